// APoTQuantRCF_91285234909889
// MI455X (gfx1250) — compile-verified
//
#include <hip/hip_runtime.h>

// APoT (BITS=5, K=2) fake-quant, elementwise, fp32 -> fp32.
// Memory-bound streaming kernel: B128 NT loads/stores + global_prefetch_b8.
// Bank0 levels {0, 1, 1/4, 1/16}  -> midpoints 0.03125, 0.15625, 0.625
// Bank1 levels {0, 1/2, 1/8, 1/32}-> midpoints 0.015625, 0.078125, 0.3125
// LMAX = 1.5, EPS = 1e-8

#define APOT_THREADS 256
#define APOT_IPT 8            // target grid-stride iterations per thread
#define APOT_PF_DIST 4        // prefetch distance in grid strides

// Native clang vector type — required by __builtin_nontemporal_load/store
// (HIP's float4 is a HIP_vector_type class and is rejected).
typedef float v4f __attribute__((ext_vector_type(4)));

__device__ __forceinline__ float apot_quant1(float xv, float inv_alpha, float alpha) {
  const float inv_lmax = 1.0f / (1.5f + 1e-8f);
  float a = fminf(fabsf(xv) * inv_alpha, 1.0f);

  // nearest level in bank0 (greedy step 1)
  float c0 = (a >= 0.625f)    ? 1.0f
           : (a >= 0.15625f)  ? 0.25f
           : (a >  0.03125f)  ? 0.0625f
           :                    0.0f;
  float r = fmaxf(a - c0, 0.0f);

  // nearest level in bank1 (greedy step 2)
  float c1 = (r >= 0.3125f)   ? 0.5f
           : (r >= 0.078125f) ? 0.125f
           : (r >  0.015625f) ? 0.03125f
           :                    0.0f;

  float acc = c0 + c1;
  float mag = fminf(acc * inv_lmax, 1.0f) * alpha;
  return copysignf(mag, xv);
}

__global__ __launch_bounds__(APOT_THREADS)
void APoTQuant_kernel(const float* __restrict__ x,
                      const float* __restrict__ alpha_p,
                      float* __restrict__ out,
                      long long n4, long long n) {
  const float alpha = alpha_p[0];                 // scalar, hoisted to s_load
  const float inv_alpha = 1.0f / (alpha + 1e-8f);

  const long long stride = (long long)gridDim.x * APOT_THREADS;
  long long i = (long long)blockIdx.x * APOT_THREADS + threadIdx.x;

  const v4f* __restrict__ x4 = (const v4f*)x;
  v4f* __restrict__ o4 = (v4f*)out;

  for (; i < n4; i += stride) {
    // Pull a future tile toward L2 (lowers to global_prefetch_b8;
    // speculative prefetches past the end are silently dropped).
    __builtin_prefetch((const void*)(x4 + i + stride * APOT_PF_DIST), 0, 0);

    v4f v = __builtin_nontemporal_load(&x4[i]);    // global_load_b128, NT
    v4f r;
    r.x = apot_quant1(v.x, inv_alpha, alpha);
    r.y = apot_quant1(v.y, inv_alpha, alpha);
    r.z = apot_quant1(v.z, inv_alpha, alpha);
    r.w = apot_quant1(v.w, inv_alpha, alpha);
    __builtin_nontemporal_store(r, &o4[i]);        // global_store_b128, NT
  }

  // Scalar tail for n % 4 != 0 (not hit for 8192*8192, kept for generality).
  long long tail_start = n4 * 4;
  long long t = (long long)blockIdx.x * APOT_THREADS + threadIdx.x;
  if (t < (n - tail_start)) {
    long long j = tail_start + t;
    out[j] = apot_quant1(x[j], inv_alpha, alpha);
  }
}

extern "C" void kernel_launch(void* const* d_in, const int* in_sizes, int n_in,
                              void* d_out, int out_size, void* d_ws, size_t ws_size,
                              hipStream_t stream) {
  (void)n_in; (void)d_ws; (void)ws_size; (void)out_size;

  const float* x     = (const float*)d_in[0];   // [8192*8192] fp32
  const float* alpha = (const float*)d_in[1];   // scalar fp32
  float* out = (float*)d_out;

  long long n  = (long long)in_sizes[0];
  long long n4 = n / 4;

  long long want = (n4 + (long long)APOT_THREADS * APOT_IPT - 1) /
                   ((long long)APOT_THREADS * APOT_IPT);
  if (want < 1) want = 1;
  if (want > 1048576) want = 1048576;
  int blocks = (int)want;

  APoTQuant_kernel<<<blocks, APOT_THREADS, 0, stream>>>(x, alpha, out, n4, n);
}